// FeatureTokenizer_47742856462564
// MI455X (gfx1250) — compile-verified
//
#include <hip/hip_runtime.h>
#include <stdint.h>

typedef float    v4f   __attribute__((ext_vector_type(4)));
typedef uint32_t u32;
typedef u32      u32x4 __attribute__((ext_vector_type(4)));
typedef int      i32x4 __attribute__((ext_vector_type(4)));
typedef int      i32x8 __attribute__((ext_vector_type(8)));

#define BDIM 8192
#define FDIM 200
#define DDIM 128

#if __has_builtin(__builtin_amdgcn_tensor_load_to_lds) && __has_builtin(__builtin_amdgcn_s_wait_tensorcnt)
#define HAVE_TDM 1
#else
#define HAVE_TDM 0
#endif

// LDS byte offset of a __shared__ object: generic (flat) shared pointers are
// {shared_aperture_hi32, lds_offset_lo32}, so truncation gives the LDS offset.
static __device__ __forceinline__ u32 lds_byte_offset(const void* p) {
  return (u32)(uintptr_t)p;
}

#if HAVE_TDM
// TDM: load one contiguous row of 128 fp32 (512B) from global into LDS.
// D# layout per CDNA5 ISA ch.8 (group0 128b, group1 256b; groups 2/3 zero —
// dims 2..4 unused, tile_dim1=0 => 1-D tile). Verified: lowers to
// `tensor_load_to_lds s[a:a+3], s[b:b+7]` on this toolchain.
static __device__ __forceinline__ void tdm_load_row_f32x128(const float* gsrc, u32 lds_off) {
  uint64_t ga = (uint64_t)(uintptr_t)gsrc;
  u32x4 g0;
  g0.x = 1u;                                             // count=1 (valid user D#), is_restore=0, gather=0
  g0.y = lds_off;                                        // lds_addr (bytes)
  g0.z = (u32)(ga & 0xFFFFFFFFu);                        // global_addr[31:0]
  g0.w = (u32)((ga >> 32) & 0x1FFFFFFu) | (2u << 30);    // global_addr[56:32] | type=2 ("image")
  i32x8 g1;
  g1[0] = (2 << 16);            // workgroup_mask=0 (not in cluster), data_size=2 (4B), no pad/iterate
  g1[1] = (int)(128u << 16);    // tensor_dim0[15:0]=128 at bits[63:48]
  g1[2] = (int)(1u   << 16);    // tensor_dim0[31:16]=0, tensor_dim1[15:0]=1
  g1[3] = (int)(128u << 16);    // tensor_dim1[31:16]=0, tile_dim0=128
  g1[4] = 0;                    // tile_dim1=0 (unused -> 1-D), tile_dim2=0
  g1[5] = 128;                  // tensor_dim0_stride[31:0]=128 (elements)
  g1[6] = 0;                    // stride hi / tensor_dim1_stride lo
  g1[7] = 0;
  i32x4 g2 = {0, 0, 0, 0};      // tensor_dim2/3, stride2, tile_dim3 unused
  i32x4 g3 = {0, 0, 0, 0};      // dim3 stride / dim4 unused
  i32x8 g4 = {0, 0, 0, 0, 0, 0, 0, 0};  // trailing descriptor words: unused
  __builtin_amdgcn_tensor_load_to_lds(g0, g1, g2, g3, g4, 0);
}
#endif

// tokens = x*W + b ; LayerNorm over D with closed-form stats:
//   mu  = x*E[W] + E[b]
//   var = x^2*Var(W) + 2x*Cov(W,b) + Var(b)
// Block: fixed f = blockIdx.x, 256 b-rows per block (8 waves x 32 rows).
__global__ __launch_bounds__(256) void FeatureTokenizerLN_kernel(
    const float* __restrict__ x,      // (B,F)
    const float* __restrict__ W,      // (F,D)
    const float* __restrict__ bias,   // (F,D)
    const float* __restrict__ gamma,  // (D)
    const float* __restrict__ beta,   // (D)
    float* __restrict__ out)          // (B,F,D)
{
  __shared__ float sW[DDIM];
  __shared__ float sB[DDIM];

  const int f    = blockIdx.x;
  const int tid  = threadIdx.x;
  const int lane = tid & 31;
  const int wave = tid >> 5;

#if HAVE_TDM
  if (wave == 0) {               // one wave issues both TDM descriptors
    tdm_load_row_f32x128(W    + (size_t)f * DDIM, lds_byte_offset(sW));
    tdm_load_row_f32x128(bias + (size_t)f * DDIM, lds_byte_offset(sB));
    __builtin_amdgcn_s_wait_tensorcnt(0);   // LDS visible before the barrier
  }
#else
  if (tid < DDIM) {
    sW[tid] = W[(size_t)f * DDIM + tid];
    sB[tid] = bias[(size_t)f * DDIM + tid];
  }
#endif
  __syncthreads();

  // Register copies: lane holds 4 consecutive d (ds_load_b128).
  const v4f wf  = *(const v4f*)&sW[lane * 4];
  const v4f bf  = *(const v4f*)&sB[lane * 4];
  const v4f gm4 = *(const v4f*)&gamma[lane * 4];
  const v4f bt4 = *(const v4f*)&beta[lane * 4];

  // Per-f moments (each wave computes its own copy; 5-step shfl_xor tree).
  float sw  = wf.x + wf.y + wf.z + wf.w;
  float sb  = bf.x + bf.y + bf.z + bf.w;
  float sww = wf.x*wf.x + wf.y*wf.y + wf.z*wf.z + wf.w*wf.w;
  float swb = wf.x*bf.x + wf.y*bf.y + wf.z*bf.z + wf.w*bf.w;
  float sbb = bf.x*bf.x + bf.y*bf.y + bf.z*bf.z + bf.w*bf.w;
#pragma unroll
  for (int off = 16; off > 0; off >>= 1) {
    sw  += __shfl_xor(sw,  off, 32);
    sb  += __shfl_xor(sb,  off, 32);
    sww += __shfl_xor(sww, off, 32);
    swb += __shfl_xor(swb, off, 32);
    sbb += __shfl_xor(sbb, off, 32);
  }
  const float inv = 1.0f / (float)DDIM;
  const float wm  = sw * inv;
  const float bm  = sb * inv;
  const float cww = sww * inv - wm * wm;   // Var(W_f)
  const float cwb = swb * inv - wm * bm;   // Cov(W_f, b_f)
  const float cbb = sbb * inv - bm * bm;   // Var(b_f)

  // 32 consecutive b-rows per wave; lane i preloads x for row b0+i.
  const int b0 = blockIdx.y * 256 + wave * 32;
  const float xl = x[(size_t)(b0 + lane) * FDIM + f];
  const int xli  = __builtin_bit_cast(int, xl);

  float* outp = out + ((size_t)b0 * FDIM + f) * DDIM + (size_t)lane * 4;
  const size_t row_stride = (size_t)FDIM * DDIM;   // floats between consecutive b

  // Fully unrolled: readlane with immediate lane index -> SGPR broadcast
  // (no ds_bpermute in the hot loop; LDS pipe stays idle, stores dominate).
#pragma unroll
  for (int j = 0; j < 32; ++j) {
    const float xj   = __builtin_bit_cast(float, __builtin_amdgcn_readlane(xli, j));
    const float mu   = xj * wm + bm;
    const float var  = (xj * xj) * cww + 2.0f * xj * cwb + cbb;
    const float rstd = __builtin_amdgcn_rsqf(var + 1e-5f);
    v4f o = ((xj * wf + bf) - mu) * rstd * gm4 + bt4;
    // Write-once 839MB output: non-temporal b128 store, fully coalesced per wave.
    __builtin_nontemporal_store(o, (v4f*)(outp + (size_t)j * row_stride));
  }
}

extern "C" void kernel_launch(void* const* d_in, const int* in_sizes, int n_in,
                              void* d_out, int out_size, void* d_ws, size_t ws_size,
                              hipStream_t stream) {
  const float* x     = (const float*)d_in[0];
  const float* W     = (const float*)d_in[1];
  const float* bias  = (const float*)d_in[2];
  const float* gamma = (const float*)d_in[3];
  const float* beta  = (const float*)d_in[4];
  float* out = (float*)d_out;

  dim3 grid(FDIM, BDIM / 256);   // 200 x 32 blocks
  dim3 block(256);               // 8 wave32
  FeatureTokenizerLN_kernel<<<grid, block, 0, stream>>>(x, W, bias, gamma, beta, out);
}